// BallPooling_72292889526275
// MI455X (gfx1250) — compile-verified
//
#include <hip/hip_runtime.h>

typedef float v2f __attribute__((ext_vector_type(2)));
typedef float v8f __attribute__((ext_vector_type(8)));

#define EPS 1e-5f

// D(16x16 f32) = A(16x4 f32) x B(4x16 f32) + C
__device__ __forceinline__ v8f wmma4(v2f a, v2f b, v8f c) {
  return __builtin_amdgcn_wmma_f32_16x16x4_f32(
      /*neg_a=*/false, a, /*neg_b=*/false, b,
      /*c_mod=*/(short)0, c, /*reuse_a=*/false, /*reuse_b=*/false);
}

// LDS layout (dword offsets):
//   WG   [k=8][grade=5][lane=32][slot=2]  : 2560  @ 0      (w_mv fragments)
//   WM2S [k=8][lane=32][slot=2]           : 512   @ 2560   (w_mv2s fragments)
//   WS   [kk=9][lane=32][slot=2]          : 576   @ 3072   (w_s fragments, K padded 34->36)
//   WS2  [kk=9][lane=32][slot=2]          : 576   @ 3648   (w_s2mv fragments)
//   BS   [16]                             : 16    @ 4224
//   ASC  [wave=4][kk=9][lane=32][slot=2]  : 2304  @ 4240   (sc_cat fragments per wave)
#define L_WG   0
#define L_WM2S 2560
#define L_WS   3072
#define L_WS2  3648
#define L_BS   4224
#define L_ASC  4240
#define L_TOT  6544

__global__ __launch_bounds__(128) void ball_pool_kernel(
    const float* __restrict__ mv, const float* __restrict__ sc,
    const float* __restrict__ pos, const float* __restrict__ w_mv,
    const float* __restrict__ w_s2mv, const float* __restrict__ w_mv2s,
    const float* __restrict__ w_s, const float* __restrict__ b_s,
    const long long* __restrict__ batch_idx,
    float* __restrict__ out_mv, float* __restrict__ out_sc,
    float* __restrict__ out_cen, long long* __restrict__ out_batch)
{
  __shared__ float lds[L_TOT];
  const int tid  = threadIdx.x;
  const int wave = tid >> 5;
  const int lane = tid & 31;
  const int hi   = lane >> 4;     // half-wave select (K offset +2 / M offset +8)
  const int gl   = lane & 15;     // N index (o) in B/C/D fragments; M index (group) in A
  const int gbase = blockIdx.x * 64 + wave * 16;  // first group of this wave's tile

  // ---- cooperative weight staging into fragment-order LDS ----
  // B fragment element (lane l, slot s) holds B[K = 4k + s + 2*(l>>4), N = l&15].
  for (int e = tid; e < 2560; e += 128) {          // w_mv (O=16, 32, 5)
    int s = e & 1, l = (e >> 1) & 31, gr = (e >> 6) % 5, k = e / 320;
    int o = l & 15, i = 4 * k + s + 2 * (l >> 4);
    lds[L_WG + e] = w_mv[(o * 32 + i) * 5 + gr];
  }
  for (int e = tid; e < 512; e += 128) {           // w_mv2s (16, 32)
    int s = e & 1, l = (e >> 1) & 31, k = e >> 6;
    int o = l & 15, i = 4 * k + s + 2 * (l >> 4);
    lds[L_WM2S + e] = w_mv2s[o * 32 + i];
  }
  for (int e = tid; e < 576; e += 128) {           // w_s (16, 34) padded to K=36
    int s = e & 1, l = (e >> 1) & 31, kk = e >> 6;
    int o = l & 15, j = 4 * kk + s + 2 * (l >> 4);
    lds[L_WS + e]  = (j < 34) ? w_s[o * 34 + j] : 0.0f;
  }
  for (int e = tid; e < 576; e += 128) {           // w_s2mv (16, 34) padded
    int s = e & 1, l = (e >> 1) & 31, kk = e >> 6;
    int o = l & 15, j = 4 * kk + s + 2 * (l >> 4);
    lds[L_WS2 + e] = (j < 34) ? w_s2mv[o * 34 + j] : 0.0f;
  }
  if (tid < 16) lds[L_BS + tid] = b_s[tid];

  // ---- per-wave: sc_cat fragment staging (coalesced 512-float tile) ----
  {
    const float4* scp = (const float4*)(sc + (size_t)gbase * 32 + (size_t)lane * 16);
    float v[16];
#pragma unroll
    for (int q4 = 0; q4 < 4; ++q4) {
      float4 t = scp[q4];
      v[4 * q4] = t.x; v[4 * q4 + 1] = t.y; v[4 * q4 + 2] = t.z; v[4 * q4 + 3] = t.w;
    }
    const int asc = L_ASC + wave * 576;
    const int g_loc = lane >> 1;
#pragma unroll
    for (int q = 0; q < 16; ++q) {
      int j = (lane & 1) * 16 + q;          // column in sc_cat
      int kk = j >> 2, c = j & 3;
      int l2 = g_loc + ((c >= 2) ? 16 : 0);
      lds[asc + (kk * 32 + l2) * 2 + (c & 1)] = v[q];
    }
  }

  // ---- per-group geometry: centers, pair distance, batch passthrough (lanes 0-15) ----
  if (lane < 16) {
    const int g = gbase + lane;
    const float* pp = pos + (size_t)(2 * g) * 3;
    float p0x = pp[0], p0y = pp[1], p0z = pp[2];
    float p1x = pp[3], p1y = pp[4], p1z = pp[5];
    float cx = 0.5f * (p0x + p1x), cy = 0.5f * (p0y + p1y), cz = 0.5f * (p0z + p1z);
    out_cen[(size_t)g * 3 + 0] = cx;
    out_cen[(size_t)g * 3 + 1] = cy;
    out_cen[(size_t)g * 3 + 2] = cz;
    float dx = p1x - p0x, dy = p1y - p0y, dz = p1z - p0z;
    float dist = 0.5f * sqrtf(dx * dx + dy * dy + dz * dz);  // same for both members
    const int asc = L_ASC + wave * 576;
    // sc_cat columns 32,33 = rel_dist; 34,35 = zero pad   (kk = 8)
    lds[asc + (8 * 32 + lane) * 2 + 0] = dist;        // j=32
    lds[asc + (8 * 32 + lane) * 2 + 1] = dist;        // j=33
    lds[asc + (8 * 32 + lane + 16) * 2 + 0] = 0.0f;   // j=34
    lds[asc + (8 * 32 + lane + 16) * 2 + 1] = 0.0f;   // j=35
    out_batch[g] = batch_idx[2 * g];
  }
  __syncthreads();

  // ---- main contraction: acc[y] (16x16 f32) over 8 K-chunks of 4 ----
  v8f acc[16];
  v8f accsc = {};
#pragma unroll
  for (int y = 0; y < 16; ++y) acc[y] = (v8f){};

  constexpr int GIDX[16] = {0,1,1,1,1,2,2,2,2,2,2,3,3,3,3,4};

  // A fragment source: lane supplies rows i0 = 4k + 2*hi and i0+1 of group gl.
  const size_t mvrow = ((size_t)(gbase + gl) * 32 + 2 * hi) * 16;

#pragma unroll
  for (int k = 0; k < 8; ++k) {
    const float4* ap = (const float4*)(mv + mvrow + (size_t)(4 * k) * 16);
    float rr[32];                     // two contiguous 16-float rows (128B coalesced)
#pragma unroll
    for (int q = 0; q < 8; ++q) {
      float4 t = ap[q];
      rr[4 * q] = t.x; rr[4 * q + 1] = t.y; rr[4 * q + 2] = t.z; rr[4 * q + 3] = t.w;
    }
    v2f bg[5];
#pragma unroll
    for (int gr = 0; gr < 5; ++gr)
      bg[gr] = *(const v2f*)&lds[L_WG + ((k * 5 + gr) * 32 + lane) * 2];
    v2f bm = *(const v2f*)&lds[L_WM2S + (k * 32 + lane) * 2];

#pragma unroll
    for (int y = 0; y < 16; ++y) {
      v2f A = {rr[y], rr[16 + y]};
      acc[y] = wmma4(A, bg[GIDX[y]], acc[y]);
      if (y == 0) accsc = wmma4(A, bm, accsc);   // mv_r[:,:,0] @ w_mv2s.T
    }
  }

  // ---- scalar-channel WMMAs: sc_cat @ w_s.T  and  sc_cat @ w_s2mv.T (into acc[0]) ----
  {
    const int asc = L_ASC + wave * 576;
#pragma unroll
    for (int kk = 0; kk < 9; ++kk) {
      v2f A  = *(const v2f*)&lds[asc + (kk * 32 + lane) * 2];
      v2f Bs = *(const v2f*)&lds[L_WS  + (kk * 32 + lane) * 2];
      v2f B2 = *(const v2f*)&lds[L_WS2 + (kk * 32 + lane) * 2];
      accsc  = wmma4(A, Bs, accsc);
      acc[0] = wmma4(A, B2, acc[0]);
    }
  }

  // ---- bias + LayerNorm on scalar channel; C/D element (r,lane) = (g=r+8*hi, o=gl) ----
  {
    float bo = lds[L_BS + gl];
#pragma unroll
    for (int r = 0; r < 8; ++r) {
      float v = accsc[r] + bo;
      float s = v;
      s += __shfl_xor(s, 1); s += __shfl_xor(s, 2);
      s += __shfl_xor(s, 4); s += __shfl_xor(s, 8);   // sum over 16-lane half (over o)
      float mu = s * 0.0625f;
      float e = v - mu;
      float q = e * e;
      q += __shfl_xor(q, 1); q += __shfl_xor(q, 2);
      q += __shfl_xor(q, 4); q += __shfl_xor(q, 8);
      float var = q * 0.0625f;
      out_sc[(size_t)(gbase + r + 8 * hi) * 16 + gl] = e * rsqrtf(var + EPS);
    }
  }

  // ---- RMS factor over (o,y) and normalized mv store (64B contiguous per (g,o)) ----
#pragma unroll
  for (int r = 0; r < 8; ++r) {
    float sq = 0.0f;
#pragma unroll
    for (int y = 0; y < 16; ++y) sq += acc[y][r] * acc[y][r];
    sq += __shfl_xor(sq, 1); sq += __shfl_xor(sq, 2);
    sq += __shfl_xor(sq, 4); sq += __shfl_xor(sq, 8);  // sum over o
    float inv = rsqrtf(sq * 0.0625f + EPS);            // 1/factor[g]
    float4* op = (float4*)(out_mv + ((size_t)(gbase + r + 8 * hi) * 16 + gl) * 16);
#pragma unroll
    for (int p = 0; p < 4; ++p) {
      float4 w;
      w.x = acc[4 * p + 0][r] * inv;
      w.y = acc[4 * p + 1][r] * inv;
      w.z = acc[4 * p + 2][r] * inv;
      w.w = acc[4 * p + 3][r] * inv;
      op[p] = w;
    }
  }
}

extern "C" void kernel_launch(void* const* d_in, const int* in_sizes, int n_in,
                              void* d_out, int out_size, void* d_ws, size_t ws_size,
                              hipStream_t stream) {
  const float*     mv      = (const float*)d_in[0];
  const float*     sc      = (const float*)d_in[1];
  const float*     pos     = (const float*)d_in[2];
  const float*     w_mv    = (const float*)d_in[3];
  const float*     w_s2mv  = (const float*)d_in[4];
  const float*     w_mv2s  = (const float*)d_in[5];
  const float*     w_s     = (const float*)d_in[6];
  const float*     b_s     = (const float*)d_in[7];
  const long long* bidx    = (const long long*)d_in[8];

  const int N = in_sizes[0] / 256;   // (N, 16, 16) f32
  const int n = N / 2;

  float* out    = (float*)d_out;
  float* o_mv   = out;                                   // (n,16,16)
  float* o_sc   = o_mv + (size_t)n * 256;                // (n,16)
  float* o_cen  = o_sc + (size_t)n * 16;                 // (n,3)
  long long* o_b = (long long*)(o_cen + (size_t)n * 3);  // (n,) int64 tail

  const int blocks = n / 64;  // 4 waves/block, 16 groups/wave
  ball_pool_kernel<<<blocks, 128, 0, stream>>>(
      mv, sc, pos, w_mv, w_s2mv, w_mv2s, w_s, b_s, bidx,
      o_mv, o_sc, o_cen, o_b);
}